// MEW_7653631722161
// MI455X (gfx1250) — compile-verified
//
#include <hip/hip_runtime.h>

// Problem geometry (fixed by the reference).
#define B_    16
#define C_    256
#define H_    128
#define W_    128
#define HW_   (H_ * W_)
#define CHUNK_ 64

// Conv-kernel tiling.
#define ROWS      16           // output rows per block
#define LDSROWS   (ROWS + 2)   // + halo rows
#define LDSTRIDE  136          // floats per LDS row; interior at float idx 4 => byte 16 => 16B aligned

// True vector type (usable with nontemporal builtins, 16B aligned).
typedef __attribute__((ext_vector_type(4))) float v4f;

// ---------------------------------------------------------------------------
// CDNA5 async global->LDS helpers (ASYNCcnt-tracked DMA path, gfx1250).
// ---------------------------------------------------------------------------
static __device__ __forceinline__ void async_load_b128(const float* lds_dst_generic,
                                                       const float* gsrc) {
  // Per the CDNA5 aperture layout, the low 32 bits of a generic LDS pointer
  // are the LDS byte offset — exactly what VDST of the async op wants.
  unsigned lds_addr = (unsigned)(unsigned long long)(const void*)lds_dst_generic;
  unsigned long long gaddr = (unsigned long long)(const void*)gsrc;
  // VDST = per-lane LDS byte address, VADDR = per-lane 64-bit global address (GV mode).
  asm volatile("global_load_async_to_lds_b128 %0, %1, off"
               :
               : "v"(lds_addr), "v"(gaddr)
               : "memory");
}

static __device__ __forceinline__ void wait_async0() {
#if __has_builtin(__builtin_amdgcn_s_wait_asynccnt)
  __builtin_amdgcn_s_wait_asynccnt(0);
#else
  asm volatile("s_wait_asynccnt 0" ::: "memory");
#endif
}

// ---------------------------------------------------------------------------
// Kernel A: chunks 0..2 (192 channels). out = x * (1 + w[c]).
// blockIdx.y = (b*192 + c)  -> channel math + weight pick are scalar (SALU +
// s_load), each lane is just b128 NT load -> 4x v_mul -> b128 NT store.
// ---------------------------------------------------------------------------
__global__ __launch_bounds__(256) void mew_scale_kernel(const float* __restrict__ x,
                                                        const float* __restrict__ w_hw,
                                                        const float* __restrict__ w_cw,
                                                        const float* __restrict__ w_ch,
                                                        float* __restrict__ out) {
  const int plane = blockIdx.y;        // 0..3071 == b*192 + c  (block-uniform)
  const int c     = plane % 192;
  const int b     = plane / 192;

  // Block-uniform weight select -> scalar branch + scalar load.
  float w;
  if (c < CHUNK_)            w = w_hw[c];
  else if (c < 2 * CHUNK_)   w = w_cw[c - CHUNK_];
  else                       w = w_ch[c - 2 * CHUNK_];
  const float s = 1.0f + w;            // scale + residual fused

  const int i4  = blockIdx.x * 256 + threadIdx.x;          // 0..4095 (float4 within plane)
  const int off = (b * C_ + c) * (HW_ / 4) + i4;           // float4 units

  v4f v = __builtin_nontemporal_load((const v4f*)x + off);
  v *= s;
  __builtin_nontemporal_store(v, (v4f*)out + off);
}

// ---------------------------------------------------------------------------
// Kernel B: chunk 3 (channels 192..255). out = dwconv3x3(x4) + bias + x4.
// One block = one (b, c, 16-row band). Input band (18 rows incl. halo) is
// DMA'd into LDS with global_load_async_to_lds_b128, then each thread
// computes a float4 of outputs from the LDS stencil.
// ---------------------------------------------------------------------------
__global__ __launch_bounds__(256) void mew_dwconv_kernel(const float* __restrict__ x,
                                                         const float* __restrict__ dw_w,
                                                         const float* __restrict__ dw_b,
                                                         float* __restrict__ out) {
  __shared__ float lds[LDSROWS * LDSTRIDE];  // 18*136*4 = 9792 B

  const int t    = threadIdx.x;
  const int lane = t & 31;   // lane within wave32
  const int wv   = t >> 5;   // wave id 0..7 (wave-uniform)

  int blk  = blockIdx.x;
  const int BANDS = H_ / ROWS;           // 8
  int band = blk % BANDS;
  int c    = (blk / BANDS) % CHUNK_;
  int b    = blk / (BANDS * CHUNK_);

  const int imgbase = (b * C_ + (3 * CHUNK_ + c)) * HW_;  // max ~67M, fits int
  const int h0 = band * ROWS;

  // Zero the whole tile (halo columns/rows stay zero => branch-free stencil).
  for (int i = t; i < LDSROWS * LDSTRIDE; i += 256) lds[i] = 0.0f;
  __syncthreads();  // ds stores visible before async DMA overwrites interior

  // Async DMA: 18 rows (valid ones), one row per wave per iteration.
  // Row = 128 floats = 32 lanes x b128. Branch conditions are wave-uniform,
  // so EXEC is all-ones when the async op issues.
  #pragma unroll
  for (int it = 0; it < 3; ++it) {
    int r  = it * 8 + wv;          // LDS row slot 0..23 (only 0..17 valid)
    int gh = h0 - 1 + r;           // global row (-1 .. h0+16)
    if (r < LDSROWS && gh >= 0 && gh < H_) {
      const float* ldst = &lds[r * LDSTRIDE + 4 + lane * 4];  // 16B aligned
      const float* gsrc = x + imgbase + gh * W_ + lane * 4;
      async_load_b128(ldst, gsrc);
    }
  }
  wait_async0();     // drain this wave's ASYNCcnt
  __syncthreads();   // make all waves' DMA'd data visible

  // 3x3 weights + bias (c is block-uniform -> scalar-cached loads).
  float wgt[9];
  #pragma unroll
  for (int i = 0; i < 9; ++i) wgt[i] = dw_w[c * 9 + i];
  const float bias = dw_b[c];

  const int w0 = lane * 4;  // output columns w0..w0+3
  #pragma unroll
  for (int it = 0; it < 2; ++it) {
    const int ro = it * 8 + wv;        // output row in band, 0..15
    const int gh = h0 + ro;
    const float* __restrict__ row0 = &lds[(ro + 0) * LDSTRIDE + 4];
    const float* __restrict__ row1 = &lds[(ro + 1) * LDSTRIDE + 4];
    const float* __restrict__ row2 = &lds[(ro + 2) * LDSTRIDE + 4];

    v4f res;
    #pragma unroll
    for (int o = 0; o < 4; ++o) {
      const int w = w0 + o;
      float acc = bias;
      acc += row0[w - 1] * wgt[0] + row0[w] * wgt[1] + row0[w + 1] * wgt[2];
      acc += row1[w - 1] * wgt[3] + row1[w] * wgt[4] + row1[w + 1] * wgt[5];
      acc += row2[w - 1] * wgt[6] + row2[w] * wgt[7] + row2[w + 1] * wgt[8];
      acc += row1[w];  // residual (+x)
      res[o] = acc;
    }
    *(v4f*)(out + imgbase + gh * W_ + w0) = res;
  }
}

// ---------------------------------------------------------------------------
extern "C" void kernel_launch(void* const* d_in, const int* in_sizes, int n_in,
                              void* d_out, int out_size, void* d_ws, size_t ws_size,
                              hipStream_t stream) {
  (void)in_sizes; (void)n_in; (void)d_ws; (void)ws_size; (void)out_size;

  const float* x    = (const float*)d_in[0];
  const float* w_hw = (const float*)d_in[1];
  const float* w_cw = (const float*)d_in[2];
  const float* w_ch = (const float*)d_in[3];
  const float* dw_w = (const float*)d_in[4];
  const float* dw_b = (const float*)d_in[5];
  float* out = (float*)d_out;

  // Chunks 0..2: grid = (plane blocks, b*192+c planes).
  dim3 gridA(HW_ / 4 / 256, B_ * 192);   // (16, 3072)
  mew_scale_kernel<<<gridA, 256, 0, stream>>>(x, w_hw, w_cw, w_ch, out);

  // Chunk 3: one block per (b, c, 16-row band).
  const int blocksB = B_ * CHUNK_ * (H_ / ROWS);  // 8192
  mew_dwconv_kernel<<<blocksB, 256, 0, stream>>>(x, dw_w, dw_b, out);
}